// GAT1_14070312861882
// MI455X (gfx1250) — compile-verified
//
#include <hip/hip_runtime.h>
#include <math.h>

typedef float v2f __attribute__((ext_vector_type(2)));
typedef float v8f __attribute__((ext_vector_type(8)));

#define NEG_SLOPE 0.2f

// ---- monotonic float <-> uint mapping so atomicMax(u32) == float max ----
__device__ __forceinline__ unsigned f2ord(float f) {
  unsigned u = __float_as_uint(f);
  return (u & 0x80000000u) ? ~u : (u | 0x80000000u);
}
__device__ __forceinline__ float ord2f(unsigned o) {
  unsigned u = (o & 0x80000000u) ? (o ^ 0x80000000u) : ~o;
  return __uint_as_float(u);
}

// =====================================================================
// GEMM1: H[N,256] = X[N,128] @ W[128,256], fp32 WMMA 16x16x4.
// One wave owns one 16x16 output tile.  A: lane&15 = M row, (lane>>4)*2
// selects the K pair held in the two A VGPRs.  B mirrors with lane&15 = N.
// C/D: VGPR r -> row m0 + r + (lane>=16 ? 8 : 0), col n0 + (lane&15).
// =====================================================================
__global__ void gat_gemm1_wmma(const float* __restrict__ X,
                               const float* __restrict__ W,
                               float* __restrict__ H, int Mtiles) {
  int lane = threadIdx.x & 31;
  int wave = threadIdx.x >> 5;
  int tile = blockIdx.x * (blockDim.x >> 5) + wave;
  if (tile >= Mtiles * 16) return;
  int mt = tile >> 4;
  int nt = tile & 15;
  int m0 = mt << 4, n0 = nt << 4;
  int ml = lane & 15;
  int koff = (lane >> 4) << 1;       // 0 for lanes 0-15, 2 for lanes 16-31
  const float* xr = X + (size_t)(m0 + ml) * 128;
  const float* wc = W + n0 + ml;
  v8f acc = {};
#pragma unroll 4
  for (int k = 0; k < 128; k += 4) {
    v2f a, b;
    a.x = xr[k + koff];
    a.y = xr[k + koff + 1];
    b.x = wc[(size_t)(k + koff) * 256];
    b.y = wc[(size_t)(k + koff + 1) * 256];
    acc = __builtin_amdgcn_wmma_f32_16x16x4_f32(false, a, false, b,
                                                (short)0, acc, false, false);
  }
  int col = n0 + ml;
  int r0 = m0 + ((lane >> 4) << 3);
#pragma unroll
  for (int r = 0; r < 8; ++r) H[(size_t)(r0 + r) * 256 + col] = acc[r];
}

// GEMM2: G[N,16] = Hin[N,256] @ W[256,16]; single 16-wide N tile.
__global__ void gat_gemm2_wmma(const float* __restrict__ Hin,
                               const float* __restrict__ W,
                               float* __restrict__ G, int Mtiles) {
  int lane = threadIdx.x & 31;
  int wave = threadIdx.x >> 5;
  int tile = blockIdx.x * (blockDim.x >> 5) + wave;
  if (tile >= Mtiles) return;
  int m0 = tile << 4;
  int ml = lane & 15;
  int koff = (lane >> 4) << 1;
  const float* xr = Hin + (size_t)(m0 + ml) * 256;
  const float* wc = W + ml;
  v8f acc = {};
#pragma unroll 4
  for (int k = 0; k < 256; k += 4) {
    v2f a, b;
    a.x = xr[k + koff];
    a.y = xr[k + koff + 1];
    b.x = wc[(size_t)(k + koff) * 16];
    b.y = wc[(size_t)(k + koff + 1) * 16];
    acc = __builtin_amdgcn_wmma_f32_16x16x4_f32(false, a, false, b,
                                                (short)0, acc, false, false);
  }
  int r0 = m0 + ((lane >> 4) << 3);
#pragma unroll
  for (int r = 0; r < 8; ++r) G[(size_t)(r0 + r) * 16 + ml] = acc[r];
}

// =====================================================================
// Layer-1 attention coefficients per node + init of max/denominator.
// =====================================================================
__global__ void gat_att1(const float* __restrict__ H,
                         const float* __restrict__ att_s,
                         const float* __restrict__ att_d,
                         float* __restrict__ asrc, float* __restrict__ adst,
                         unsigned* __restrict__ mord, float* __restrict__ den,
                         int N) {
  int n = blockIdx.x * blockDim.x + threadIdx.x;
  if (n >= N) return;
  const float* hr = H + (size_t)n * 256;
#pragma unroll
  for (int h = 0; h < 8; ++h) {
    float s = 0.f, d = 0.f;
#pragma unroll
    for (int c = 0; c < 32; ++c) {
      float v = hr[h * 32 + c];
      s = fmaf(v, att_s[h * 32 + c], s);
      d = fmaf(v, att_d[h * 32 + c], d);
    }
    asrc[n * 8 + h] = s;
    adst[n * 8 + h] = d;
    mord[n * 8 + h] = 0u;   // ord-encoded -inf
    den [n * 8 + h] = 0.f;
  }
}

__device__ __forceinline__ void edge_sd(const int* __restrict__ ei, int E,
                                        int e, int& s, int& d) {
  if (e < E) { s = ei[e]; d = ei[E + e]; } else { s = d = e - E; }
}

__global__ void gat_edge_max1(const int* __restrict__ ei, int E, int Etot,
                              const float* __restrict__ asrc,
                              const float* __restrict__ adst,
                              unsigned* __restrict__ mord) {
  int e = blockIdx.x * blockDim.x + threadIdx.x;
  if (e >= Etot) return;
  int s, d; edge_sd(ei, E, e, s, d);
#pragma unroll
  for (int h = 0; h < 8; ++h) {
    float v = asrc[s * 8 + h] + adst[d * 8 + h];
    v = v > 0.f ? v : NEG_SLOPE * v;
    atomicMax(&mord[d * 8 + h], f2ord(v));
  }
}

__global__ void gat_edge_sum1(const int* __restrict__ ei, int E, int Etot,
                              const float* __restrict__ asrc,
                              const float* __restrict__ adst,
                              const unsigned* __restrict__ mord,
                              float* __restrict__ den) {
  int e = blockIdx.x * blockDim.x + threadIdx.x;
  if (e >= Etot) return;
  int s, d; edge_sd(ei, E, e, s, d);
#pragma unroll
  for (int h = 0; h < 8; ++h) {
    float v = asrc[s * 8 + h] + adst[d * 8 + h];
    v = v > 0.f ? v : NEG_SLOPE * v;
    atomicAdd(&den[d * 8 + h], __expf(v - ord2f(mord[d * 8 + h])));
  }
}

__global__ void gat_bias_init(float* __restrict__ out,
                              const float* __restrict__ bias,
                              int total, int Cmask) {
  int i = blockIdx.x * blockDim.x + threadIdx.x;
  if (i >= total) return;
  out[i] = bias[i & Cmask];
}

// Heavy kernel: wave per edge, lane handles 8 contiguous channels
// (head = lane/4).  2x float4 gather from h[src], 8 atomic adds to out[dst].
__global__ void gat_edge_aggr1(const int* __restrict__ ei, int E, int Etot,
                               const float* __restrict__ H,
                               const float* __restrict__ asrc,
                               const float* __restrict__ adst,
                               const unsigned* __restrict__ mord,
                               const float* __restrict__ den,
                               float* __restrict__ out) {
  int gid = blockIdx.x * blockDim.x + threadIdx.x;
  int e = gid >> 5;
  int lane = threadIdx.x & 31;
  if (e >= Etot) return;
  int s, d; edge_sd(ei, E, e, s, d);
  int h = lane >> 2;
  float v = asrc[s * 8 + h] + adst[d * 8 + h];
  v = v > 0.f ? v : NEG_SLOPE * v;
  float alpha = __expf(v - ord2f(mord[d * 8 + h])) / den[d * 8 + h];
  const float4* hp = (const float4*)(H + (size_t)s * 256 + lane * 8);
  float4 v0 = hp[0], v1 = hp[1];
  float* op = out + (size_t)d * 256 + lane * 8;
  atomicAdd(op + 0, v0.x * alpha); atomicAdd(op + 1, v0.y * alpha);
  atomicAdd(op + 2, v0.z * alpha); atomicAdd(op + 3, v0.w * alpha);
  atomicAdd(op + 4, v1.x * alpha); atomicAdd(op + 5, v1.y * alpha);
  atomicAdd(op + 6, v1.z * alpha); atomicAdd(op + 7, v1.w * alpha);
}

__global__ void gat_elu(float* __restrict__ x, int total) {
  int i = blockIdx.x * blockDim.x + threadIdx.x;
  if (i >= total) return;
  float v = x[i];
  x[i] = v > 0.f ? v : expm1f(v);
}

// =====================================================================
// Layer 2 (H=1, C=16)
// =====================================================================
__global__ void gat_att2(const float* __restrict__ G,
                         const float* __restrict__ att_s,
                         const float* __restrict__ att_d,
                         float* __restrict__ asrc, float* __restrict__ adst,
                         unsigned* __restrict__ mord, float* __restrict__ den,
                         int N) {
  int n = blockIdx.x * blockDim.x + threadIdx.x;
  if (n >= N) return;
  const float* g = G + (size_t)n * 16;
  float s = 0.f, d = 0.f;
#pragma unroll
  for (int c = 0; c < 16; ++c) {
    s = fmaf(g[c], att_s[c], s);
    d = fmaf(g[c], att_d[c], d);
  }
  asrc[n] = s; adst[n] = d; mord[n] = 0u; den[n] = 0.f;
}

__global__ void gat_edge_max2(const int* __restrict__ ei, int E, int Etot,
                              const float* __restrict__ asrc,
                              const float* __restrict__ adst,
                              unsigned* __restrict__ mord) {
  int e = blockIdx.x * blockDim.x + threadIdx.x;
  if (e >= Etot) return;
  int s, d; edge_sd(ei, E, e, s, d);
  float v = asrc[s] + adst[d];
  v = v > 0.f ? v : NEG_SLOPE * v;
  atomicMax(&mord[d], f2ord(v));
}

__global__ void gat_edge_sum2(const int* __restrict__ ei, int E, int Etot,
                              const float* __restrict__ asrc,
                              const float* __restrict__ adst,
                              const unsigned* __restrict__ mord,
                              float* __restrict__ den) {
  int e = blockIdx.x * blockDim.x + threadIdx.x;
  if (e >= Etot) return;
  int s, d; edge_sd(ei, E, e, s, d);
  float v = asrc[s] + adst[d];
  v = v > 0.f ? v : NEG_SLOPE * v;
  atomicAdd(&den[d], __expf(v - ord2f(mord[d])));
}

__global__ void gat_edge_aggr2(const int* __restrict__ ei, int E, int Etot,
                               const float* __restrict__ G,
                               const float* __restrict__ asrc,
                               const float* __restrict__ adst,
                               const unsigned* __restrict__ mord,
                               const float* __restrict__ den,
                               float* __restrict__ out) {
  int gid = blockIdx.x * blockDim.x + threadIdx.x;
  int e = gid >> 4;
  int c = gid & 15;
  if (e >= Etot) return;
  int s, d; edge_sd(ei, E, e, s, d);
  float v = asrc[s] + adst[d];
  v = v > 0.f ? v : NEG_SLOPE * v;
  float alpha = __expf(v - ord2f(mord[d])) / den[d];
  atomicAdd(&out[(size_t)d * 16 + c], G[(size_t)s * 16 + c] * alpha);
}

// =====================================================================
extern "C" void kernel_launch(void* const* d_in, const int* in_sizes, int n_in,
                              void* d_out, int out_size, void* d_ws, size_t ws_size,
                              hipStream_t stream) {
  const float* x   = (const float*)d_in[0];
  const int*   ei  = (const int*)  d_in[1];
  const float* W1  = (const float*)d_in[2];
  const float* as1 = (const float*)d_in[3];
  const float* ad1 = (const float*)d_in[4];
  const float* b1  = (const float*)d_in[5];
  const float* W2  = (const float*)d_in[6];
  const float* as2 = (const float*)d_in[7];
  const float* ad2 = (const float*)d_in[8];
  const float* b2  = (const float*)d_in[9];
  float* out = (float*)d_out;

  const int N    = in_sizes[0] / 128;   // 50000
  const int E    = in_sizes[1] / 2;     // 1000000
  const int Etot = E + N;               // + self loops
  const int Mtiles = N / 16;            // 3125 (N is a multiple of 16)

  // workspace carve-up (~113 MB)
  float*    h1    = (float*)d_ws;                          // N*256
  float*    o1    = h1 + (size_t)N * 256;                  // N*256 (out1 -> elu -> h2)
  float*    asrc1 = o1 + (size_t)N * 256;                  // N*8
  float*    adst1 = asrc1 + (size_t)N * 8;                 // N*8
  unsigned* m1    = (unsigned*)(adst1 + (size_t)N * 8);    // N*8
  float*    den1  = (float*)(m1 + (size_t)N * 8);          // N*8
  float*    g2    = den1 + (size_t)N * 8;                  // N*16
  float*    asrc2 = g2 + (size_t)N * 16;                   // N
  float*    adst2 = asrc2 + N;                             // N
  unsigned* m2    = (unsigned*)(adst2 + N);                // N
  float*    den2  = (float*)(m2 + N);                      // N

  // ---- layer 1 ----
  {
    int tiles = Mtiles * 16;
    gat_gemm1_wmma<<<(tiles + 7) / 8, 256, 0, stream>>>(x, W1, h1, Mtiles);
  }
  gat_att1<<<(N + 255) / 256, 256, 0, stream>>>(h1, as1, ad1, asrc1, adst1, m1, den1, N);
  gat_edge_max1<<<(Etot + 255) / 256, 256, 0, stream>>>(ei, E, Etot, asrc1, adst1, m1);
  gat_edge_sum1<<<(Etot + 255) / 256, 256, 0, stream>>>(ei, E, Etot, asrc1, adst1, m1, den1);
  gat_bias_init<<<(N * 256 + 255) / 256, 256, 0, stream>>>(o1, b1, N * 256, 255);
  {
    long long threads = (long long)Etot * 32;
    gat_edge_aggr1<<<(int)((threads + 255) / 256), 256, 0, stream>>>(
        ei, E, Etot, h1, asrc1, adst1, m1, den1, o1);
  }
  gat_elu<<<(N * 256 + 255) / 256, 256, 0, stream>>>(o1, N * 256);

  // ---- layer 2 ----
  gat_gemm2_wmma<<<(Mtiles + 7) / 8, 256, 0, stream>>>(o1, W2, g2, Mtiles);
  gat_att2<<<(N + 255) / 256, 256, 0, stream>>>(g2, as2, ad2, asrc2, adst2, m2, den2, N);
  gat_edge_max2<<<(Etot + 255) / 256, 256, 0, stream>>>(ei, E, Etot, asrc2, adst2, m2);
  gat_edge_sum2<<<(Etot + 255) / 256, 256, 0, stream>>>(ei, E, Etot, asrc2, adst2, m2, den2);
  gat_bias_init<<<(N * 16 + 255) / 256, 256, 0, stream>>>(out, b2, N * 16, 15);
  {
    long long threads = (long long)Etot * 16;
    gat_edge_aggr2<<<(int)((threads + 255) / 256), 256, 0, stream>>>(
        ei, E, Etot, g2, asrc2, adst2, m2, den2, out);
  }
}